// OptNet_1640677507400
// MI455X (gfx1250) — compile-verified
//
#include <hip/hip_runtime.h>
#include <math.h>

#define N_OBS     25
#define M_ROWS    31          // 1 CLF + 25 CBF + 5 fixed rows
#define IPM_ITERS 25
#define SIGMA     0.1f
#define U_LIM     0.05f
#define P_SLACK   0.05f

#if defined(__has_builtin)
#if __has_builtin(__builtin_amdgcn_global_load_async_to_lds_b32)
#define HAVE_ASYNC_LDS 1
#endif
#endif

typedef __attribute__((address_space(1))) int* gint_p;   // global (LangAS device)
typedef __attribute__((address_space(3))) int* lint_p;   // LDS

__global__ __launch_bounds__(256) void optnet_ipm_kernel(
    const float* __restrict__ x,       const float* __restrict__ xt,
    const float* __restrict__ circles, const float* __restrict__ radius,
    const float* __restrict__ lbd,     const float* __restrict__ alpha,
    float* __restrict__ out, int n_samples)
{
    // sconst: [0..49]=circles(xy), [50..74]=radius, [75]=lbd, [76]=alpha
    __shared__ float sconst[2 * N_OBS + N_OBS + 2];
    const int t = threadIdx.x;

#ifdef HAVE_ASYNC_LDS
    // gfx1250 async global->LDS copy (ASYNCcnt), one dword per enabled lane.
    if (t < 77) {
        const float* g = (t < 50) ? (circles + t)
                       : (t < 75) ? (radius + (t - 50))
                       : (t == 75) ? lbd : alpha;
        __builtin_amdgcn_global_load_async_to_lds_b32(
            (gint_p)g, (lint_p)&sconst[t], 0, 0);
    }
    asm volatile("s_wait_asynccnt 0" ::: "memory");
#else
    if (t < 77) {
        sconst[t] = (t < 50) ? circles[t]
                  : (t < 75) ? radius[t - 50]
                  : (t == 75) ? lbd[0] : alpha[0];
    }
#endif
    __syncthreads();

    const int n = blockIdx.x * blockDim.x + t;
    if (n >= n_samples) return;

    const float lb = sconst[75];
    const float al = sconst[76];

    const float x0 = x[2 * n],  x1 = x[2 * n + 1];
    const float y0 = xt[2 * n], y1 = xt[2 * n + 1];

    // ---- Build G (rows: gx,gy,gz) and h once; fully unrolled -> registers ----
    float gxr[M_ROWS], gyr[M_ROWS], gzr[M_ROWS], hr[M_ROWS];
    {
        const float dx0 = x0 - y0, dx1 = x1 - y1;
        gxr[0] = 2.0f * dx0; gyr[0] = 2.0f * dx1; gzr[0] = 1.0f;
        hr[0]  = -lb * (dx0 * dx0 + dx1 * dx1);
    }
#pragma unroll
    for (int o = 0; o < N_OBS; ++o) {
        const float cx = sconst[2 * o], cy = sconst[2 * o + 1];
        const float ddx = x0 - cx, ddy = x1 - cy;
        const float nr  = sqrtf(ddx * ddx + ddy * ddy);
        const float inv = __fdividef(1.0f, nr + 1e-6f);
        gxr[1 + o] = -ddx * inv;
        gyr[1 + o] = -ddy * inv;
        gzr[1 + o] = -1.0f;
        hr[1 + o]  = al * (nr - sconst[50 + o]);
    }
    gxr[26] =  1.0f; gyr[26] =  0.0f; gzr[26] =  0.0f; hr[26] = U_LIM;
    gxr[27] = -1.0f; gyr[27] =  0.0f; gzr[27] =  0.0f; hr[27] = U_LIM;
    gxr[28] =  0.0f; gyr[28] =  1.0f; gzr[28] =  0.0f; hr[28] = U_LIM;
    gxr[29] =  0.0f; gyr[29] = -1.0f; gzr[29] =  0.0f; hr[29] = U_LIM;
    gxr[30] =  0.0f; gyr[30] =  0.0f; gzr[30] = -1.0f; hr[30] = 0.0f;

    float z0 = 0.0f, z1 = 0.0f, z2 = 0.0f;
    float sv[M_ROWS], lv[M_ROWS];
#pragma unroll
    for (int j = 0; j < M_ROWS; ++j) { sv[j] = 1.0f; lv[j] = 1.0f; }

#pragma unroll 1
    for (int it = 0; it < IPM_ITERS; ++it) {
        // duality measure
        float mu = 0.0f;
#pragma unroll
        for (int j = 0; j < M_ROWS; ++j) mu += sv[j] * lv[j];
        const float smu = SIGMA * (mu * (1.0f / (float)M_ROWS));

        // M = Q + G' D G  (6 uniques), rhs = -(Qz + G'(lam + w))
        float m00 = 1.0f, m01 = 0.0f, m02 = 0.0f;
        float m11 = 1.0f, m12 = 0.0f, m22 = P_SLACK;
        float r0 = z0, r1 = z1, r2 = P_SLACK * z2;   // Qz (Q diagonal)
        float dv[M_ROWS], wv[M_ROWS];
#pragma unroll
        for (int j = 0; j < M_ROWS; ++j) {
            const float rs = __fdividef(1.0f, sv[j]);      // 1/s
            const float dd = lv[j] * rs;                   // d = lam/s
            const float gdot = gxr[j] * z0 + gyr[j] * z1 + gzr[j] * z2;
            const float rp = gdot + sv[j] - hr[j];
            const float w  = dd * rp + lv[j] - smu * rs;
            dv[j] = dd; wv[j] = w;
            const float c = lv[j] + w;
            r0 += gxr[j] * c;  r1 += gyr[j] * c;  r2 += gzr[j] * c;
            const float dgx = dd * gxr[j], dgy = dd * gyr[j];
            m00 += dgx * gxr[j];  m01 += dgx * gyr[j];  m02 += dgx * gzr[j];
            m11 += dgy * gyr[j];  m12 += dgy * gzr[j];  m22 += dd * gzr[j] * gzr[j];
        }
        r0 = -r0; r1 = -r1; r2 = -r2;

        // 3x3 SPD solve via adjugate (symmetric)
        const float c00 = m11 * m22 - m12 * m12;
        const float c01 = m02 * m12 - m01 * m22;
        const float c02 = m01 * m12 - m02 * m11;
        const float c11 = m00 * m22 - m02 * m02;
        const float c12 = m01 * m02 - m00 * m12;
        const float c22 = m00 * m11 - m01 * m01;
        const float det  = m00 * c00 + m01 * c01 + m02 * c02;
        const float idet = __fdividef(1.0f, det);
        const float dz0 = (c00 * r0 + c01 * r1 + c02 * r2) * idet;
        const float dz1 = (c01 * r0 + c11 * r1 + c12 * r2) * idet;
        const float dz2 = (c02 * r0 + c12 * r1 + c22 * r2) * idet;

        // dlam, ds; fraction-to-boundary via max of ratios:
        //   min_{ds<0}(-s/ds)  ==  1 / max_j(-ds * (1/s))   (ratios <=0 ignored)
        // reuses 1/lam (needed for ds) and 1/s = d * (1/lam) -> no per-row extra rcp
        float ms = 0.0f, ml = 0.0f;
        float dlv[M_ROWS], dsv[M_ROWS];
#pragma unroll
        for (int j = 0; j < M_ROWS; ++j) {
            const float gdz  = gxr[j] * dz0 + gyr[j] * dz1 + gzr[j] * dz2;
            const float dlam = dv[j] * gdz + wv[j];
            const float rl   = __fdividef(1.0f, lv[j]);    // 1/lam
            const float ds   = -(sv[j] * lv[j] - smu + sv[j] * dlam) * rl;
            dlv[j] = dlam; dsv[j] = ds;
            const float rs   = dv[j] * rl;                 // (lam/s)*(1/lam) = 1/s
            ms = fmaxf(ms, -ds * rs);
            ml = fmaxf(ml, -dlam * rl);
        }
        const float mm = fmaxf(ms, ml);
        const float aa = (mm > 0.0f)
                       ? fminf(1.0f, 0.99f * __fdividef(1.0f, mm))
                       : 1.0f;

        z0 += aa * dz0; z1 += aa * dz1; z2 += aa * dz2;
#pragma unroll
        for (int j = 0; j < M_ROWS; ++j) {
            sv[j] += aa * dsv[j];
            lv[j] += aa * dlv[j];
        }
    }

    out[3 * n + 0] = z0;
    out[3 * n + 1] = z1;
    out[3 * n + 2] = z2;
}

extern "C" void kernel_launch(void* const* d_in, const int* in_sizes, int n_in,
                              void* d_out, int out_size, void* d_ws, size_t ws_size,
                              hipStream_t stream) {
    const float* x       = (const float*)d_in[0];
    const float* xt      = (const float*)d_in[1];
    const float* circles = (const float*)d_in[2];
    const float* radius  = (const float*)d_in[3];
    const float* lbd     = (const float*)d_in[4];
    const float* alpha   = (const float*)d_in[5];
    float* out = (float*)d_out;

    const int n = in_sizes[0] / 2;
    dim3 block(256);
    dim3 grid((n + 255) / 256);
    hipLaunchKernelGGL(optnet_ipm_kernel, grid, block, 0, stream,
                       x, xt, circles, radius, lbd, alpha, out, n);
    (void)n_in; (void)out_size; (void)d_ws; (void)ws_size;
}